// GraphEncoder_335007449146
// MI455X (gfx1250) — compile-verified
//
#include <hip/hip_runtime.h>

typedef float v2f __attribute__((ext_vector_type(2)));
typedef float v8f __attribute__((ext_vector_type(8)));

#define N_NODES  50000
#define N_GRAPHS 64
#define D_IN     64
#define D_H      128
#define N_EDGES  600000

static __host__ int cdiv(int n, int b) { return (n + b - 1) / b; }

// ---------- elementwise helpers ----------
__global__ void k_fill(float* __restrict__ p, float v, int n) {
  int i = blockIdx.x * blockDim.x + threadIdx.x;
  if (i < n) p[i] = v;
}

__global__ void k_count(const int* __restrict__ dst, float* __restrict__ cnt, int E) {
  int e = blockIdx.x * blockDim.x + threadIdx.x;
  if (e < E) atomicAdd(&cnt[dst[e]], 1.0f);
}

__global__ void k_recip_clamp(float* __restrict__ c, int n) {
  int i = blockIdx.x * blockDim.x + threadIdx.x;
  if (i < n) c[i] = 1.0f / fmaxf(c[i], 1.0f);
}

// one thread per (edge, feature); coalesced over features
__global__ void k_scatter_add(const int* __restrict__ src, const int* __restrict__ dst,
                              const float* __restrict__ x, float* __restrict__ agg,
                              int E, int dshift) {
  int idx = blockIdx.x * blockDim.x + threadIdx.x;
  int total = E << dshift;
  if (idx >= total) return;
  int e = idx >> dshift;
  int f = idx & ((1 << dshift) - 1);
  atomicAdd(&agg[(dst[e] << dshift) + f], x[(src[e] << dshift) + f]);
}

__global__ void k_scale_rows(float* __restrict__ agg, const float* __restrict__ inv,
                             int total, int dshift) {
  int i = blockIdx.x * blockDim.x + threadIdx.x;
  if (i < total) agg[i] *= inv[i >> dshift];
}

// ---------- fused SAGE GEMM: O = [relu](A@Wl + X@Wr + bl), N = D_H = 128 ----------
// One wave32 computes one 16x16 tile of O via V_WMMA_F32_16X16X4_F32.
// A fragment: lane m = lane&15 holds A[m][k0+2h], A[m][k0+2h+1], h = lane>>4.
// B fragment mirrored: lane n = lane&15 holds W[k0+2h][n], W[k0+2h+1][n].
// Acc v8f: VGPR r -> (M = r + 8h, N = lane&15).
template <int KDIM, bool RELU>
__global__ void __launch_bounds__(256)
k_sage_gemm(const float* __restrict__ A,   // [M, KDIM] aggregated, pre-scaled
            const float* __restrict__ X,   // [M, KDIM] self features
            const float* __restrict__ Wl,  // [KDIM, D_H]
            const float* __restrict__ Wr,  // [KDIM, D_H]
            const float* __restrict__ bl,  // [D_H]
            float* __restrict__ O) {       // [M, D_H]
  int wave  = (blockIdx.x * blockDim.x + threadIdx.x) >> 5;
  int lane  = threadIdx.x & 31;
  int mtile = wave >> 3;          // 8 N-tiles per M-tile
  int ntile = wave & 7;
  int half  = lane >> 4;          // 0 or 1
  int l16   = lane & 15;

  int row = mtile * 16 + l16;     // A/X row held by this lane
  int col = ntile * 16 + l16;     // output / weight column held by this lane

  float bias = bl[col];
  v8f acc = {bias, bias, bias, bias, bias, bias, bias, bias};

  const float* arow = A + row * KDIM;
  const float* xrow = X + row * KDIM;

#pragma unroll 4
  for (int k0 = 0; k0 < KDIM; k0 += 4) {
    int k = k0 + 2 * half;
    v2f a, b;
    // neighbor-mean term: agg @ Wl
    a.x = arow[k];
    a.y = arow[k + 1];
    b.x = Wl[k * D_H + col];
    b.y = Wl[(k + 1) * D_H + col];
    acc = __builtin_amdgcn_wmma_f32_16x16x4_f32(false, a, false, b, (short)0, acc,
                                                false, false);
    // self term: x @ Wr
    a.x = xrow[k];
    a.y = xrow[k + 1];
    b.x = Wr[k * D_H + col];
    b.y = Wr[(k + 1) * D_H + col];
    acc = __builtin_amdgcn_wmma_f32_16x16x4_f32(false, a, false, b, (short)0, acc,
                                                false, false);
  }

  int obase = (mtile * 16 + half * 8) * D_H + col;
#pragma unroll
  for (int r = 0; r < 8; ++r) {
    float v = acc[r];
    if (RELU) v = fmaxf(v, 0.0f);
    O[obase + r * D_H] = v;
  }
}

// ---------- global mean pool ----------
__global__ void k_pool_cnt(const int* __restrict__ batch, float* __restrict__ pcnt, int n) {
  int i = blockIdx.x * blockDim.x + threadIdx.x;
  if (i < n) atomicAdd(&pcnt[batch[i]], 1.0f);
}

__global__ void k_pool_sum(const int* __restrict__ batch, const float* __restrict__ h,
                           float* __restrict__ psum, int n) {
  int idx = blockIdx.x * blockDim.x + threadIdx.x;
  if (idx >= n * D_H) return;
  int node = idx >> 7;
  int f    = idx & (D_H - 1);
  atomicAdd(&psum[batch[node] * D_H + f], h[idx]);
}

__global__ void k_pool_div(const float* __restrict__ psum, const float* __restrict__ pcnt,
                           float* __restrict__ out) {
  int idx = blockIdx.x * blockDim.x + threadIdx.x;
  if (idx >= N_GRAPHS * D_H) return;
  out[idx] = psum[idx] / fmaxf(pcnt[idx >> 7], 1.0f);
}

extern "C" void kernel_launch(void* const* d_in, const int* in_sizes, int n_in,
                              void* d_out, int out_size, void* d_ws, size_t ws_size,
                              hipStream_t stream) {
  const float* x     = (const float*)d_in[0];
  const int*   ei    = (const int*)d_in[1];     // [2, E]
  const int*   src   = ei;
  const int*   dst   = ei + N_EDGES;
  const int*   batch = (const int*)d_in[2];
  const float* Wl1 = (const float*)d_in[3];
  const float* bl1 = (const float*)d_in[4];
  const float* Wr1 = (const float*)d_in[5];
  const float* Wl2 = (const float*)d_in[6];
  const float* bl2 = (const float*)d_in[7];
  const float* Wr2 = (const float*)d_in[8];
  const float* Wl3 = (const float*)d_in[9];
  const float* bl3 = (const float*)d_in[10];
  const float* Wr3 = (const float*)d_in[11];
  float* out = (float*)d_out;

  // workspace layout (floats)
  float* cnt  = (float*)d_ws;                 // N_NODES (inverse degree after k_recip_clamp)
  float* agg  = cnt + 50048;                  // N_NODES * D_H (256B aligned)
  float* hA   = agg + N_NODES * D_H;          // N_NODES * D_H
  float* hB   = hA + N_NODES * D_H;           // N_NODES * D_H
  float* psum = hB + N_NODES * D_H;           // N_GRAPHS * D_H
  float* pcnt = psum + N_GRAPHS * D_H;        // N_GRAPHS

  const int B = 256;
  const int gemm_blocks = (N_NODES / 16) * 8 / 8;  // 3125 blocks x 8 waves = 25000 tiles

  // degrees (shared by all 3 layers) + pool accumulators
  k_fill<<<cdiv(N_NODES, B), B, 0, stream>>>(cnt, 0.0f, N_NODES);
  k_fill<<<cdiv(N_GRAPHS * D_H + N_GRAPHS, B), B, 0, stream>>>(psum, 0.0f,
                                                               N_GRAPHS * D_H + N_GRAPHS);
  k_count<<<cdiv(N_EDGES, B), B, 0, stream>>>(dst, cnt, N_EDGES);
  k_recip_clamp<<<cdiv(N_NODES, B), B, 0, stream>>>(cnt, N_NODES);

  // ---- layer 1 (K = 64, ReLU) ----
  k_fill<<<cdiv(N_NODES * D_IN, B), B, 0, stream>>>(agg, 0.0f, N_NODES * D_IN);
  k_scatter_add<<<cdiv(N_EDGES * D_IN, B), B, 0, stream>>>(src, dst, x, agg, N_EDGES, 6);
  k_scale_rows<<<cdiv(N_NODES * D_IN, B), B, 0, stream>>>(agg, cnt, N_NODES * D_IN, 6);
  k_sage_gemm<D_IN, true><<<gemm_blocks, B, 0, stream>>>(agg, x, Wl1, Wr1, bl1, hA);

  // ---- layer 2 (K = 128, ReLU) ----
  k_fill<<<cdiv(N_NODES * D_H, B), B, 0, stream>>>(agg, 0.0f, N_NODES * D_H);
  k_scatter_add<<<cdiv(N_EDGES * D_H, B), B, 0, stream>>>(src, dst, hA, agg, N_EDGES, 7);
  k_scale_rows<<<cdiv(N_NODES * D_H, B), B, 0, stream>>>(agg, cnt, N_NODES * D_H, 7);
  k_sage_gemm<D_H, true><<<gemm_blocks, B, 0, stream>>>(agg, hA, Wl2, Wr2, bl2, hB);

  // ---- layer 3 (K = 128, no ReLU) ----
  k_fill<<<cdiv(N_NODES * D_H, B), B, 0, stream>>>(agg, 0.0f, N_NODES * D_H);
  k_scatter_add<<<cdiv(N_EDGES * D_H, B), B, 0, stream>>>(src, dst, hB, agg, N_EDGES, 7);
  k_scale_rows<<<cdiv(N_NODES * D_H, B), B, 0, stream>>>(agg, cnt, N_NODES * D_H, 7);
  k_sage_gemm<D_H, false><<<gemm_blocks, B, 0, stream>>>(agg, hB, Wl3, Wr3, bl3, hA);

  // ---- global mean pool ----
  k_pool_cnt<<<cdiv(N_NODES, B), B, 0, stream>>>(batch, pcnt, N_NODES);
  k_pool_sum<<<cdiv(N_NODES * D_H, B), B, 0, stream>>>(batch, hA, psum, N_NODES);
  k_pool_div<<<cdiv(N_GRAPHS * D_H, B), B, 0, stream>>>(psum, pcnt, out);
}